// Model_83150566850959
// MI455X (gfx1250) — compile-verified
//
#include <hip/hip_runtime.h>
#include <math.h>

// Problem constants (match reference)
#define BB 16
#define NN 16384
#define LL 192
#define EE 128
#define DD 128

typedef __attribute__((ext_vector_type(2))) float v2f;
typedef __attribute__((ext_vector_type(4))) float v4f;
typedef __attribute__((ext_vector_type(8))) float v8f;

// D = A(16x4 f32) * B(4x16 f32) + C(16x16 f32)  — CDNA5 V_WMMA_F32_16X16X4_F32
__device__ __forceinline__ v8f wmma_f32_4(v2f a, v2f b, v8f c) {
  return __builtin_amdgcn_wmma_f32_16x16x4_f32(false, a, false, b, (short)0, c,
                                               false, false);
}

// ---------------------------------------------------------------------------
// obs = relu(feats @ W_obs + b_obs) * mask      feats = [x, 1-mask+ymask]
// one block = 8 nodes, each of 32 lanes per node writes a float4 of D
// ---------------------------------------------------------------------------
__global__ void obs_kernel(const float* __restrict__ x,
                           const float* __restrict__ mask,
                           const float* __restrict__ ymask,
                           const float* __restrict__ Wobs,   // [2, D]
                           const float* __restrict__ bobs,   // [D]
                           float* __restrict__ obs) {        // [B*N, D]
  const int node = blockIdx.x * 8 + (threadIdx.x >> 5);
  const int d4 = (threadIdx.x & 31) * 4;
  const float xv = x[node];
  const float mv = mask[node];
  const float f1 = 1.0f - mv + ymask[node];
  const v4f w0 = *(const v4f*)(Wobs + d4);
  const v4f w1 = *(const v4f*)(Wobs + DD + d4);
  const v4f bb = *(const v4f*)(bobs + d4);
  v4f r;
  r.x = fmaxf(fmaf(xv, w0.x, fmaf(f1, w1.x, bb.x)), 0.0f) * mv;
  r.y = fmaxf(fmaf(xv, w0.y, fmaf(f1, w1.y, bb.y)), 0.0f) * mv;
  r.z = fmaxf(fmaf(xv, w0.z, fmaf(f1, w1.z, bb.z)), 0.0f) * mv;
  r.w = fmaxf(fmaf(xv, w0.w, fmaf(f1, w1.w, bb.w)), 0.0f) * mv;
  __builtin_nontemporal_store(r, (v4f*)(obs + (size_t)node * DD + d4));
}

// ---------------------------------------------------------------------------
// ti[b,l,n] = (time_idx[b,n]==l)*mask ; vi[b,e,n] = (var_idx[b,n]==e)*mask
// block covers 1024 columns n of one batch; streams NT float4 stores
// ---------------------------------------------------------------------------
__global__ void incidence_kernel(const float* __restrict__ mask,
                                 const int* __restrict__ vidx,
                                 const int* __restrict__ tidx,
                                 float* __restrict__ ti,   // [B*L, N]
                                 float* __restrict__ vi) { // [B*E, N]
  const int b = blockIdx.x / (NN / 1024);
  const int n0 = (blockIdx.x % (NN / 1024)) * 1024 + threadIdx.x * 4;
  const size_t base = (size_t)b * NN + n0;
  const int4 tv = *(const int4*)(tidx + base);
  const int4 vv = *(const int4*)(vidx + base);
  const v4f mv = *(const v4f*)(mask + base);
#pragma unroll 4
  for (int l = 0; l < LL; ++l) {
    v4f r;
    r.x = (tv.x == l) ? mv.x : 0.0f;
    r.y = (tv.y == l) ? mv.y : 0.0f;
    r.z = (tv.z == l) ? mv.z : 0.0f;
    r.w = (tv.w == l) ? mv.w : 0.0f;
    __builtin_nontemporal_store(
        r, (v4f*)(ti + (size_t)(b * LL + l) * NN + n0));
  }
#pragma unroll 4
  for (int e = 0; e < EE; ++e) {
    v4f r;
    r.x = (vv.x == e) ? mv.x : 0.0f;
    r.y = (vv.y == e) ? mv.y : 0.0f;
    r.z = (vv.z == e) ? mv.z : 0.0f;
    r.w = (vv.w == e) ? mv.w : 0.0f;
    __builtin_nontemporal_store(
        r, (v4f*)(vi + (size_t)(b * EE + e) * NN + n0));
  }
}

// ---------------------------------------------------------------------------
// temp_he[b,l,d] = sin(mark[b,l] * W_time[d] + b_time[d])
// ---------------------------------------------------------------------------
__global__ void temp_kernel(const float* __restrict__ mark,  // [B*L]
                            const float* __restrict__ Wt,    // [D]
                            const float* __restrict__ bt,    // [D]
                            float* __restrict__ out) {       // [B*L, D]
  const int idx = blockIdx.x * blockDim.x + threadIdx.x;
  const int d = idx & (DD - 1);
  const int bl = idx >> 7;
  out[idx] = sinf(fmaf(mark[bl], Wt[d], bt[d]));
}

// ---------------------------------------------------------------------------
// cnt[b,e] = sum over masked nodes with var e of mask^2  (== att_aux diag * sqrtN)
// ---------------------------------------------------------------------------
__global__ void zero_cnt_kernel(float* __restrict__ cnt) {
  cnt[blockIdx.x * blockDim.x + threadIdx.x] = 0.0f;
}

__global__ void count_kernel(const float* __restrict__ mask,
                             const int* __restrict__ vidx,
                             float* __restrict__ cnt) {  // [B, E]
  const int node = blockIdx.x * 256 + threadIdx.x;
  const float m = mask[node];
  if (m != 0.0f) {
    const int b = node / NN;
    atomicAdd(&cnt[b * EE + vidx[node]], m * m);
  }
}

// ---------------------------------------------------------------------------
// q/k/v = relu(var_hyperedge_w) @ W + b   (batch-independent, 128x128x128 f32)
// grid.x = 3 (q,k,v).  8 waves, each owns a 16-row strip; WMMA f32 16x16x4.
// ---------------------------------------------------------------------------
__global__ void qkv_kernel(const float* __restrict__ vhw,  // [E, D]
                           const float* __restrict__ Wq, const float* __restrict__ bq,
                           const float* __restrict__ Wk, const float* __restrict__ bk,
                           const float* __restrict__ Wv, const float* __restrict__ bv,
                           float* __restrict__ qg, float* __restrict__ kg,
                           float* __restrict__ vg) {
  __shared__ float sA[EE * DD];  // 64 KB: relu(var_hyperedge_w)
  for (int i = threadIdx.x; i < EE * DD / 4; i += 256) {
    v4f t = *(const v4f*)(vhw + i * 4);
    t.x = fmaxf(t.x, 0.0f);
    t.y = fmaxf(t.y, 0.0f);
    t.z = fmaxf(t.z, 0.0f);
    t.w = fmaxf(t.w, 0.0f);
    *(v4f*)(sA + i * 4) = t;
  }
  __syncthreads();

  const float* W;
  const float* bias;
  float* outp;
  if (blockIdx.x == 0)      { W = Wq; bias = bq; outp = qg; }
  else if (blockIdx.x == 1) { W = Wk; bias = bk; outp = kg; }
  else                      { W = Wv; bias = bv; outp = vg; }

  const int wave = threadIdx.x >> 5;
  const int lane = threadIdx.x & 31;
  const int r = lane & 15;
  const int half = lane >> 4;
  const int m0 = wave * 16;

  for (int n0 = 0; n0 < DD; n0 += 16) {
    v8f acc = {};
#pragma unroll 8
    for (int k0 = 0; k0 < DD; k0 += 4) {
      v2f a, bf;
      a.x = sA[(m0 + r) * DD + k0 + 2 * half];
      a.y = sA[(m0 + r) * DD + k0 + 2 * half + 1];
      bf.x = W[(k0 + 2 * half) * DD + n0 + r];
      bf.y = W[(k0 + 2 * half + 1) * DD + n0 + r];
      acc = wmma_f32_4(a, bf, acc);
    }
    const float bb = bias[n0 + r];
#pragma unroll
    for (int i = 0; i < 8; ++i)
      outp[(size_t)(m0 + i + 8 * half) * DD + n0 + r] = acc[i] + bb;
  }
}

// ---------------------------------------------------------------------------
// att0 = (q @ k^T) / sqrt(D)   (batch-independent 128x128)
// ---------------------------------------------------------------------------
__global__ void att_kernel(const float* __restrict__ qg,
                           const float* __restrict__ kg,
                           float* __restrict__ att0) {  // [E, E]
  const int wave = threadIdx.x >> 5;
  const int lane = threadIdx.x & 31;
  const int r = lane & 15;
  const int half = lane >> 4;
  const int m0 = wave * 16;
  const float scale = 0.08838834764831845f;  // 1/sqrt(128)

  for (int n0 = 0; n0 < EE; n0 += 16) {
    v8f acc = {};
#pragma unroll 8
    for (int k0 = 0; k0 < DD; k0 += 4) {
      v2f a, bf;
      a.x = qg[(m0 + r) * DD + k0 + 2 * half];
      a.y = qg[(m0 + r) * DD + k0 + 2 * half + 1];
      bf.x = kg[(n0 + r) * DD + k0 + 2 * half];      // B = k^T
      bf.y = kg[(n0 + r) * DD + k0 + 2 * half + 1];
      acc = wmma_f32_4(a, bf, acc);
    }
#pragma unroll
    for (int i = 0; i < 8; ++i)
      att0[(size_t)(m0 + i + 8 * half) * EE + n0 + r] = acc[i] * scale;
  }
}

// ---------------------------------------------------------------------------
// per batch: diagonal merge (att_aux is exactly diagonal), row softmax,
// out = attn @ v   — one block per batch, attn kept in LDS, WMMA for the GEMM
// ---------------------------------------------------------------------------
__global__ void attn_out_kernel(const float* __restrict__ att0,
                                const float* __restrict__ vg,
                                const float* __restrict__ cnt,
                                const float* __restrict__ thr_p,
                                const float* __restrict__ mc_p,
                                const int* __restrict__ nmax_p,
                                float* __restrict__ outp) {  // [B, E, D]
  __shared__ float satt[EE * EE];  // 64 KB
  const int b = blockIdx.x;
  for (int i = threadIdx.x; i < EE * EE / 4; i += 256)
    *(v4f*)(satt + i * 4) = *(const v4f*)(att0 + i * 4);
  __syncthreads();

  if (threadIdx.x < EE) {
    const int e = threadIdx.x;
    const float thr = thr_p[0];
    const float mc = mc_p[0];
    const float aux = cnt[b * EE + e] * rsqrtf((float)nmax_p[0]);
    const float de = satt[e * EE + e];
    if (de > thr && aux != 0.0f)
      satt[e * EE + e] = (1.0f - mc) * de + mc * aux;
    // softmax over row e
    float mx = -3.402823466e38f;
    for (int f = 0; f < EE; ++f) mx = fmaxf(mx, satt[e * EE + f]);
    float s = 0.0f;
    for (int f = 0; f < EE; ++f) {
      const float ex = __expf(satt[e * EE + f] - mx);
      satt[e * EE + f] = ex;
      s += ex;
    }
    const float inv = 1.0f / s;
    for (int f = 0; f < EE; ++f) satt[e * EE + f] *= inv;
  }
  __syncthreads();

  const int wave = threadIdx.x >> 5;
  const int lane = threadIdx.x & 31;
  const int r = lane & 15;
  const int half = lane >> 4;
  const int m0 = wave * 16;
  float* ob = outp + (size_t)b * EE * DD;

  for (int n0 = 0; n0 < DD; n0 += 16) {
    v8f acc = {};
#pragma unroll 8
    for (int k0 = 0; k0 < EE; k0 += 4) {
      v2f a, bf;
      a.x = satt[(m0 + r) * EE + k0 + 2 * half];
      a.y = satt[(m0 + r) * EE + k0 + 2 * half + 1];
      bf.x = vg[(k0 + 2 * half) * DD + n0 + r];
      bf.y = vg[(k0 + 2 * half + 1) * DD + n0 + r];
      acc = wmma_f32_4(a, bf, acc);
    }
#pragma unroll
    for (int i = 0; i < 8; ++i)
      ob[(size_t)(m0 + i + 8 * half) * DD + n0 + r] = acc[i];
  }
}

// ---------------------------------------------------------------------------
extern "C" void kernel_launch(void* const* d_in, const int* in_sizes, int n_in,
                              void* d_out, int out_size, void* d_ws,
                              size_t ws_size, hipStream_t stream) {
  const float* x     = (const float*)d_in[0];
  const float* mask  = (const float*)d_in[1];
  const float* ymask = (const float*)d_in[2];
  const float* mark  = (const float*)d_in[3];
  const int*   vidx  = (const int*)d_in[4];
  const int*   tidx  = (const int*)d_in[5];
  const int*   nmax  = (const int*)d_in[6];
  const float* Wobs  = (const float*)d_in[7];
  const float* bobs  = (const float*)d_in[8];
  const float* Wt    = (const float*)d_in[9];
  const float* bt    = (const float*)d_in[10];
  const float* vhw   = (const float*)d_in[11];
  const float* Wq    = (const float*)d_in[12];
  const float* bq    = (const float*)d_in[13];
  const float* Wk    = (const float*)d_in[14];
  const float* bk    = (const float*)d_in[15];
  const float* Wv    = (const float*)d_in[16];
  const float* bv    = (const float*)d_in[17];
  const float* thr   = (const float*)d_in[18];
  const float* mc    = (const float*)d_in[19];

  // output tuple layout (flat concat): obs, temp_he, out, ti, vi
  float* out_f = (float*)d_out;
  const size_t obs_off  = 0;
  const size_t temp_off = obs_off + (size_t)BB * NN * DD;   // 33554432
  const size_t att_off  = temp_off + (size_t)BB * LL * DD;  // +393216
  const size_t ti_off   = att_off + (size_t)BB * EE * DD;   // +262144
  const size_t vi_off   = ti_off + (size_t)BB * LL * NN;    // +50331648
  float* obs = out_f + obs_off;
  float* th  = out_f + temp_off;
  float* ao  = out_f + att_off;
  float* ti  = out_f + ti_off;
  float* vi  = out_f + vi_off;

  // workspace: cnt[B*E], q/k/v[E*D], att0[E*E]
  float* ws   = (float*)d_ws;
  float* cnt  = ws;                 // 2048
  float* qg   = cnt + BB * EE;      // 16384
  float* kg   = qg + EE * DD;
  float* vg   = kg + EE * DD;
  float* att0 = vg + EE * DD;       // 16384
  (void)in_sizes; (void)n_in; (void)out_size; (void)ws_size;

  zero_cnt_kernel<<<(BB * EE) / 256, 256, 0, stream>>>(cnt);
  count_kernel<<<(BB * NN) / 256, 256, 0, stream>>>(mask, vidx, cnt);
  obs_kernel<<<(BB * NN) / 8, 256, 0, stream>>>(x, mask, ymask, Wobs, bobs, obs);
  incidence_kernel<<<BB * (NN / 1024), 256, 0, stream>>>(mask, vidx, tidx, ti, vi);
  temp_kernel<<<(BB * LL * DD) / 256, 256, 0, stream>>>(mark, Wt, bt, th);
  qkv_kernel<<<3, 256, 0, stream>>>(vhw, Wq, bq, Wk, bk, Wv, bv, qg, kg, vg);
  att_kernel<<<1, 256, 0, stream>>>(qg, kg, att0);
  attn_out_kernel<<<BB, 256, 0, stream>>>(att0, vg, cnt, thr, mc, nmax, ao);
}